// GCNLayer_39402029973984
// MI455X (gfx1250) — compile-verified
//
#include <hip/hip_runtime.h>
#include <hip/hip_bf16.h>

#define NUSER 80000
#define NITEM 40000
#define DIM   128

typedef float v2f __attribute__((ext_vector_type(2)));
typedef float v8f __attribute__((ext_vector_type(8)));

// ---------------------------------------------------------------------------
// Zero-fill (grid-stride) — accumulators must be zeroed every call.
// ---------------------------------------------------------------------------
__global__ void zero_kernel(float* __restrict__ p, size_t n) {
    size_t i = (size_t)blockIdx.x * blockDim.x + threadIdx.x;
    size_t stride = (size_t)gridDim.x * blockDim.x;
    for (; i < n; i += stride) p[i] = 0.0f;
}

// ---------------------------------------------------------------------------
// Fused bidirectional SpMM for one behavior:
//   out_u[row] += val * xi[col]   (A  @ item-side emb)
//   out_i[col] += val * xu[row]   (A^T @ user-side emb)
// One wave (32 lanes) per edge; each lane owns 4 consecutive floats of D=128.
// Gather tables fit in L2 (192MB); scatter via global_atomic_add_f32 (L2 RMW).
// ---------------------------------------------------------------------------
__global__ __launch_bounds__(256) void spmm_pair_kernel(
    const int* __restrict__ rows, const int* __restrict__ cols,
    const float* __restrict__ vals,
    const float* __restrict__ xi,   // [NITEM][DIM]
    const float* __restrict__ xu,   // [NUSER][DIM]
    float* __restrict__ out_u,      // [NUSER][DIM]
    float* __restrict__ out_i,      // [NITEM][DIM]
    int nnz)
{
    int tid  = blockIdx.x * blockDim.x + threadIdx.x;
    int e    = tid >> 5;
    if (e >= nnz) return;
    int lane = tid & 31;

    int   r = rows[e];
    int   c = cols[e];
    float v = vals[e];

    const float4 a = *(const float4*)(xi + (size_t)c * DIM + lane * 4);
    const float4 b = *(const float4*)(xu + (size_t)r * DIM + lane * 4);

    float* ou = out_u + (size_t)r * DIM + lane * 4;
    float* oi = out_i + (size_t)c * DIM + lane * 4;

    atomicAdd(ou + 0, v * a.x); atomicAdd(ou + 1, v * a.y);
    atomicAdd(ou + 2, v * a.z); atomicAdd(ou + 3, v * a.w);
    atomicAdd(oi + 0, v * b.x); atomicAdd(oi + 1, v * b.y);
    atomicAdd(oi + 2, v * b.z); atomicAdd(oi + 3, v * b.w);
}

// ---------------------------------------------------------------------------
// Column-wise sum of squares over the node axis: sumsq[d] += sum_r emb[r][d]^2
// blockDim.x == 128 (thread == feature d); one atomic per (block, d).
// ---------------------------------------------------------------------------
__global__ __launch_bounds__(128) void sumsq_kernel(
    const float* __restrict__ emb, float* __restrict__ sumsq, int n)
{
    const int ROWS = 256;
    int d  = threadIdx.x;
    int r0 = blockIdx.x * ROWS;
    int r1 = min(r0 + ROWS, n);
    float s = 0.0f;
    for (int r = r0; r < r1; ++r) {
        float x = emb[(size_t)r * DIM + d];
        s += x * x;
    }
    atomicAdd(sumsq + d, s);
}

// inv_norm = 1 / max(sqrt(sumsq), EPS) for all 8*128 columns
__global__ void invnorm_kernel(const float* __restrict__ sumsq,
                               float* __restrict__ invn) {
    int i = blockIdx.x * blockDim.x + threadIdx.x;
    if (i < 8 * DIM) {
        float nrm = sqrtf(sumsq[i]);
        invn[i] = 1.0f / fmaxf(nrm, 1e-12f);
    }
}

// out[n][d] = emb[n][d] * invn[d]   (column-wise normalize)
__global__ void scale_kernel(const float* __restrict__ emb,
                             const float* __restrict__ invn,
                             float* __restrict__ out, size_t n) {
    size_t i = (size_t)blockIdx.x * blockDim.x + threadIdx.x;
    if (i < n) out[i] = emb[i] * invn[i & (DIM - 1)];
}

// ---------------------------------------------------------------------------
// out = sigmoid( mean(e0..e3) @ W ), W is DIMxDIM (row-major, K-major rows).
// Block: 256 threads = 8 waves; block computes 16 rows x 128 cols.
// Mean tile (16x128) staged in LDS (stride 132 -> conflict-free A reads).
// Each wave owns one 16x16 C tile via V_WMMA_F32_16X16X4_F32, K-loop step 4.
//   A layout (16x4 f32): lanes 0-15 -> M=lane, VGPR0=K0, VGPR1=K1;
//                        lanes 16-31 -> M=lane-16, VGPR0=K2, VGPR1=K3.
//   B layout (4x16 f32): VGPR0: lanes0-15 K=0/N=lane, lanes16-31 K=2/N=lane-16;
//                        VGPR1: K=1 / K=3.
//   C/D (16x16 f32, 8 VGPRs): VGPR r -> M=r (lanes0-15) / M=r+8 (lanes16-31).
// ---------------------------------------------------------------------------
__global__ __launch_bounds__(256) void mean_gemm_sigmoid_wmma(
    const float* __restrict__ e0, const float* __restrict__ e1,
    const float* __restrict__ e2, const float* __restrict__ e3,
    const float* __restrict__ W,  float* __restrict__ out, int n)
{
    __shared__ float As[16][132];

    int row0 = blockIdx.x * 16;

    // Stage mean tile: 2048 elements, 256 threads, uniform 8-trip loop.
    for (int i = threadIdx.x; i < 16 * DIM; i += 256) {
        int r = i >> 7;
        int d = i & 127;
        int g = row0 + r;
        float m = 0.0f;
        if (g < n) {
            size_t idx = (size_t)g * DIM + d;
            m = 0.25f * (e0[idx] + e1[idx] + e2[idx] + e3[idx]);
        }
        As[r][d] = m;
    }
    __syncthreads();

    int wave  = threadIdx.x >> 5;
    int lane  = threadIdx.x & 31;
    int l15   = lane & 15;
    int khalf = (lane < 16) ? 0 : 2;
    int n0    = wave * 16;          // this wave's 16-column slice of W / out

    v8f c = {};
    for (int k = 0; k < DIM; k += 4) {
        v2f a, b;
        a.x = As[l15][k + khalf];
        a.y = As[l15][k + khalf + 1];
        b.x = W[(size_t)(k + khalf) * DIM + n0 + l15];
        b.y = W[(size_t)(k + khalf + 1) * DIM + n0 + l15];
        c = __builtin_amdgcn_wmma_f32_16x16x4_f32(
                /*neg_a=*/false, a, /*neg_b=*/false, b,
                /*c_mod=*/(short)0, c, /*reuse_a=*/false, /*reuse_b=*/false);
    }

    #pragma unroll
    for (int r = 0; r < 8; ++r) {
        int m = (lane < 16) ? r : r + 8;
        int g = row0 + m;
        if (g < n) {
            float x = c[r];
            out[(size_t)g * DIM + n0 + l15] = 1.0f / (1.0f + __expf(-x));
        }
    }
}

// ---------------------------------------------------------------------------
// Host-side orchestration
// ---------------------------------------------------------------------------
extern "C" void kernel_launch(void* const* d_in, const int* in_sizes, int n_in,
                              void* d_out, int out_size, void* d_ws, size_t ws_size,
                              hipStream_t stream) {
    (void)in_sizes; (void)n_in; (void)out_size; (void)ws_size;

    const float* user_embedding = (const float*)d_in[0];
    const float* item_embedding = (const float*)d_in[1];
    const float* uu_embed[3] = { (const float*)d_in[2], (const float*)d_in[4], (const float*)d_in[6] };
    const float* ii_embed[3] = { (const float*)d_in[3], (const float*)d_in[5], (const float*)d_in[7] };
    const float* u_w = (const float*)d_in[8];
    const float* i_w = (const float*)d_in[9];
    const int*   rows[4] = { (const int*)d_in[10], (const int*)d_in[13], (const int*)d_in[16], (const int*)d_in[19] };
    const int*   cols[4] = { (const int*)d_in[11], (const int*)d_in[14], (const int*)d_in[17], (const int*)d_in[20] };
    const float* vals[4] = { (const float*)d_in[12], (const float*)d_in[15], (const float*)d_in[18], (const float*)d_in[21] };
    const int nnz[4] = { in_sizes[12], in_sizes[15], in_sizes[18], in_sizes[21] };

    const size_t UD = (size_t)NUSER * DIM;
    const size_t ID = (size_t)NITEM * DIM;

    // d_out layout (tuple order): user_out, item_out, l2u[4], l2i[4], u0,i0,u1,i1,u2,i2
    float* user_out = (float*)d_out;
    float* item_out = user_out + UD;
    float* l2u      = item_out + ID;
    float* l2i      = l2u + 4 * UD;
    float* tail     = l2i + 4 * ID;   // u0,i0,u1,i1,u2,i2 (contiguous)
    float* u_acc[4];
    float* i_acc[4];
    u_acc[0] = tail;               i_acc[0] = u_acc[0] + UD;
    u_acc[1] = i_acc[0] + ID;      i_acc[1] = u_acc[1] + UD;
    u_acc[2] = i_acc[1] + ID;      i_acc[2] = u_acc[2] + UD;

    // workspace: u3, i3, sumsq[8*128], invn[8*128]
    float* ws    = (float*)d_ws;
    u_acc[3]     = ws;
    i_acc[3]     = u_acc[3] + UD;
    float* sumsq = i_acc[3] + ID;
    float* invn  = sumsq + 8 * DIM;

    // 1) zero all accumulators + sumsq (tail of d_out is contiguous u0..i2)
    zero_kernel<<<2048, 256, 0, stream>>>(tail, 3 * (UD + ID));
    zero_kernel<<<2048, 256, 0, stream>>>(u_acc[3], UD + ID + 8 * DIM);

    // 2) fused bidirectional SpMM per behavior (32 lanes per edge)
    for (int b = 0; b < 4; ++b) {
        const float* xi = (b < 3) ? ii_embed[b] : item_embedding;
        const float* xu = (b < 3) ? uu_embed[b] : user_embedding;
        long long threads = (long long)nnz[b] * 32;
        int blocks = (int)((threads + 255) / 256);
        spmm_pair_kernel<<<blocks, 256, 0, stream>>>(
            rows[b], cols[b], vals[b], xi, xu, u_acc[b], i_acc[b], nnz[b]);
    }

    // 3) column-wise sum of squares (node-axis L2 norm)
    for (int b = 0; b < 4; ++b) {
        sumsq_kernel<<<(NUSER + 255) / 256, 128, 0, stream>>>(u_acc[b], sumsq + b * DIM, NUSER);
        sumsq_kernel<<<(NITEM + 255) / 256, 128, 0, stream>>>(i_acc[b], sumsq + (4 + b) * DIM, NITEM);
    }
    invnorm_kernel<<<(8 * DIM + 255) / 256, 256, 0, stream>>>(sumsq, invn);

    // 4) write normalized stacked embeddings
    for (int b = 0; b < 4; ++b) {
        scale_kernel<<<(int)((UD + 255) / 256), 256, 0, stream>>>(
            u_acc[b], invn + b * DIM, l2u + b * UD, UD);
        scale_kernel<<<(int)((ID + 255) / 256), 256, 0, stream>>>(
            i_acc[b], invn + (4 + b) * DIM, l2i + b * ID, ID);
    }

    // 5) sigmoid(mean @ W) via f32 WMMA (16 rows x 128 cols per block)
    mean_gemm_sigmoid_wmma<<<NUSER / 16, 256, 0, stream>>>(
        u_acc[0], u_acc[1], u_acc[2], u_acc[3], u_w, user_out, NUSER);
    mean_gemm_sigmoid_wmma<<<NITEM / 16, 256, 0, stream>>>(
        i_acc[0], i_acc[1], i_acc[2], i_acc[3], i_w, item_out, NITEM);
}